// KMeansTorch_46866683134745
// MI455X (gfx1250) — compile-verified
//
#include <hip/hip_runtime.h>
#include <hip/hip_bf16.h>
#include <math.h>

// K-means: N=131072 points, D=128 dims, K=256 clusters, 8 fixed iterations.
#define NPTS 131072
#define DIM  128
#define KCL  256
#define MAXIT 8
#define TOLERANCE 1e-4f

// LDS row stride for the staged centroid matrix: 128 + 4 pad floats so that
// row stride (528 B = 132 banks mod 64 = 4) spreads the 16 lanes of each
// half across distinct banks for ds_load_b64 B-fragment reads.
#define CSTRIDE 132

typedef float v2f __attribute__((ext_vector_type(2)));
typedef float v8f __attribute__((ext_vector_type(8)));

// ---------------------------------------------------------------- init
__global__ __launch_bounds__(256) void km_init(const float* __restrict__ X,
                                               const int* __restrict__ idx,
                                               float* __restrict__ cent,
                                               int* __restrict__ done) {
    int i = blockIdx.x * 256 + threadIdx.x;           // over K*D = 32768
    int k = i >> 7;
    int d = i & (DIM - 1);
    cent[i] = X[(size_t)idx[k] * DIM + d];
    if (i == 0) *done = 0;
}

// ---------------------------------------------------------------- zero
__global__ __launch_bounds__(256) void km_zero(float* __restrict__ sums,
                                               float* __restrict__ counts,
                                               float* __restrict__ ssd) {
    int i = blockIdx.x * 256 + threadIdx.x;
    if (i < KCL * DIM) sums[i] = 0.0f;
    if (i < KCL)       counts[i] = 0.0f;
    if (i == 0)        *ssd = 0.0f;
}

// ---------------------------------------------------------------- ||c_k||^2
__global__ __launch_bounds__(256) void km_cnorm(const float* __restrict__ cent,
                                                float* __restrict__ cnorm) {
    int k = threadIdx.x;                               // one block of 256
    float s = 0.0f;
    const float* c = cent + (size_t)k * DIM;
    #pragma unroll 8
    for (int d = 0; d < DIM; ++d) { float v = c[d]; s += v * v; }
    cnorm[k] = s;
}

// ---------------------------------------------------------------- assignment (WMMA fp32)
// Block = 256 threads = 8 waves; each wave owns 32 rows (two 16-row M-tiles).
// Centroids (256x128 fp32, 128 KB) staged once per block into padded LDS;
// B fragments then come from ds_load_b64, each feeding TWO wmma ops.
//
// V_WMMA_F32_16X16X4_F32 layouts (wave32):
//   A (16x4): lanes 0-15 -> rows M=lane, K=k0+{0,1}; lanes 16-31 -> K=k0+{2,3}.
//   B (4x16): lanes 0-15 -> cols N=lane, K=k0+{0,1}; lanes 16-31 -> K=k0+{2,3}.
//   C/D 16x16: VGPR i holds row i (lanes 0-15) / row i+8 (lanes 16-31), col = lane&15.
__global__ __launch_bounds__(256) void km_assign(const float* __restrict__ X,
                                                 const float* __restrict__ cent,
                                                 const float* __restrict__ cnorm,
                                                 float* __restrict__ labels_f) {
    __shared__ float sC[KCL * CSTRIDE];                // ~132 KB of the 320 KB LDS
    __shared__ float sCn[KCL];

    // ---- stage centroids into LDS (coalesced float4 global loads) ----
    for (int g = threadIdx.x; g < KCL * DIM / 4; g += 256) {
        int row  = g >> 5;                             // 32 float4 per row
        int col4 = g & 31;
        float4 v = ((const float4*)cent)[g];
        *(float4*)&sC[row * CSTRIDE + col4 * 4] = v;   // 528B row stride: 16B aligned
    }
    if (threadIdx.x < KCL) sCn[threadIdx.x] = cnorm[threadIdx.x];
    __syncthreads();

    const int lane    = threadIdx.x & 31;
    const int wave    = threadIdx.x >> 5;
    const int l15     = lane & 15;
    const int half    = lane >> 4;                     // 0 or 1
    const int rowBase = blockIdx.x * 256 + wave * 32;  // 8 waves * 32 rows

    // ---- preload A fragments for both M-tiles (2 x 32 x v2f = 128 VGPRs) ----
    const float* x0 = X + (size_t)(rowBase + l15) * DIM + half * 2;
    const float* x1 = x0 + 16 * DIM;
    v2f a0[32], a1[32];
    #pragma unroll
    for (int kk = 0; kk < 32; ++kk) {
        a0[kk] = *(const v2f*)(x0 + kk * 4);
        a1[kk] = *(const v2f*)(x1 + kk * 4);
    }

    float bestV0[8], bestV1[8];
    int   bestI0[8], bestI1[8];
    #pragma unroll
    for (int i = 0; i < 8; ++i) {
        bestV0[i] = 3.4e38f; bestI0[i] = 0;
        bestV1[i] = 3.4e38f; bestI1[i] = 0;
    }

    for (int ct = 0; ct < 16; ++ct) {                  // 16 centroid tiles of 16
        const float* crow = sC + (ct * 16 + l15) * CSTRIDE + half * 2;
        v8f acc0 = {}, acc1 = {};
        #pragma unroll
        for (int kk = 0; kk < 32; ++kk) {
            v2f b = *(const v2f*)(crow + kk * 4);      // ds_load_b64, bank-conflict-free
            acc0 = __builtin_amdgcn_wmma_f32_16x16x4_f32(
                false, a0[kk], false, b, (short)0, acc0, false, false);
            acc1 = __builtin_amdgcn_wmma_f32_16x16x4_f32(
                false, a1[kk], false, b, (short)0, acc1, false, false);
        }
        const int   j  = ct * 16 + l15;                // this lane's column
        const float cn = sCn[j];
        #pragma unroll
        for (int i = 0; i < 8; ++i) {
            float d20 = cn - 2.0f * acc0[i];           // ||c||^2 - 2 x.c
            float d21 = cn - 2.0f * acc1[i];
            if (d20 < bestV0[i]) { bestV0[i] = d20; bestI0[i] = j; }
            if (d21 < bestV1[i]) { bestV1[i] = d21; bestI1[i] = j; }
        }
    }

    // Reduce argmin across the 16 lanes of each half (ties -> smallest index).
    #pragma unroll
    for (int m = 1; m <= 8; m <<= 1) {
        #pragma unroll
        for (int i = 0; i < 8; ++i) {
            float ov0 = __shfl_xor(bestV0[i], m, 32);
            int   oi0 = __shfl_xor(bestI0[i], m, 32);
            if (ov0 < bestV0[i] || (ov0 == bestV0[i] && oi0 < bestI0[i])) {
                bestV0[i] = ov0; bestI0[i] = oi0;
            }
            float ov1 = __shfl_xor(bestV1[i], m, 32);
            int   oi1 = __shfl_xor(bestI1[i], m, 32);
            if (ov1 < bestV1[i] || (ov1 == bestV1[i] && oi1 < bestI1[i])) {
                bestV1[i] = ov1; bestI1[i] = oi1;
            }
        }
    }
    if (l15 == 0) {                                    // lane 0 -> rows 0..7, lane 16 -> rows 8..15
        int rowOff = half * 8;
        #pragma unroll
        for (int i = 0; i < 8; ++i) {
            labels_f[rowBase + rowOff + i]      = (float)bestI0[i];
            labels_f[rowBase + 16 + rowOff + i] = (float)bestI1[i];
        }
    }
}

// ---------------------------------------------------------------- segment sums (L2 fp32 atomics)
__global__ __launch_bounds__(256) void km_accum(const float* __restrict__ X,
                                                const float* __restrict__ labels_f,
                                                float* __restrict__ sums,
                                                float* __restrict__ counts) {
    int idx = blockIdx.x * 256 + threadIdx.x;          // over N*D
    int n = idx >> 7;
    int d = idx & (DIM - 1);
    int l = (int)labels_f[n];
    atomicAdd(&sums[l * DIM + d], X[idx]);
    if (d == 0) atomicAdd(&counts[l], 1.0f);
}

// ---------------------------------------------------------------- new centroids + shift^2
__global__ __launch_bounds__(256) void km_newc(const float* __restrict__ sums,
                                               const float* __restrict__ counts,
                                               const float* __restrict__ cent,
                                               float* __restrict__ newc,
                                               float* __restrict__ ssd) {
    int idx = blockIdx.x * 256 + threadIdx.x;          // over K*D
    int k = idx >> 7;
    float v = sums[idx] / counts[k];                   // empty cluster -> NaN (matches ref)
    newc[idx] = v;
    float dlt = v - cent[idx];
    float sq  = dlt * dlt;

    __shared__ float red[256];
    red[threadIdx.x] = sq;
    __syncthreads();
    for (int s = 128; s > 0; s >>= 1) {
        if (threadIdx.x < s) red[threadIdx.x] += red[threadIdx.x + s];
        __syncthreads();
    }
    if (threadIdx.x == 0) atomicAdd(ssd, red[0]);
}

// ---------------------------------------------------------------- convergence + conditional update
__global__ __launch_bounds__(256) void km_finalize(const float* __restrict__ newc,
                                                   float* __restrict__ cent,
                                                   const float* __restrict__ ssd,
                                                   int* __restrict__ done) {
    __shared__ int sdone;
    if (threadIdx.x == 0) {
        int d = *done;
        float shift = sqrtf(*ssd);                     // NaN < TOL is false -> not done
        if (shift < TOLERANCE) d = 1;
        *done = d;
        sdone = d;
    }
    __syncthreads();
    if (!sdone) {
        for (int i = threadIdx.x; i < KCL * DIM; i += 256)
            cent[i] = newc[i];
    }
}

// ---------------------------------------------------------------- write centroids to d_out
__global__ __launch_bounds__(256) void km_copyout(const float* __restrict__ cent,
                                                  float* __restrict__ out) {
    int i = blockIdx.x * 256 + threadIdx.x;
    out[i] = cent[i];
}

extern "C" void kernel_launch(void* const* d_in, const int* in_sizes, int n_in,
                              void* d_out, int out_size, void* d_ws, size_t ws_size,
                              hipStream_t stream) {
    const float* X        = (const float*)d_in[0];
    const int*   init_idx = (const int*)d_in[1];
    // d_in[2] = max_iter (device scalar); reference fixes scan length = 8.

    float* out      = (float*)d_out;
    float* labels_f = out;             // first N elements: labels (as float)
    float* cent_out = out + NPTS;      // next K*D: centroids

    float* ws     = (float*)d_ws;      // ~395 KB of scratch
    float* cent   = ws;                // K*D
    float* newc   = ws + 32768;        // K*D
    float* sums   = ws + 65536;        // K*D
    float* counts = ws + 98304;        // K
    float* cnorm  = ws + 98560;        // K
    float* ssd    = ws + 98816;        // 1
    int*   done   = (int*)(ws + 98817);

    km_init<<<KCL * DIM / 256, 256, 0, stream>>>(X, init_idx, cent, done);

    for (int it = 0; it < MAXIT; ++it) {
        km_zero<<<(KCL * DIM + KCL + 1 + 255) / 256, 256, 0, stream>>>(sums, counts, ssd);
        km_cnorm<<<1, 256, 0, stream>>>(cent, cnorm);
        km_assign<<<NPTS / 256, 256, 0, stream>>>(X, cent, cnorm, labels_f);
        km_accum<<<NPTS * DIM / 256, 256, 0, stream>>>(X, labels_f, sums, counts);
        km_newc<<<KCL * DIM / 256, 256, 0, stream>>>(sums, counts, cent, newc, ssd);
        km_finalize<<<1, 256, 0, stream>>>(newc, cent, ssd, done);
    }

    km_copyout<<<KCL * DIM / 256, 256, 0, stream>>>(cent, cent_out);
}